// GuidedFilterHR_60576218742892
// MI455X (gfx1250) — compile-verified
//
#include <hip/hip_runtime.h>
#include <stdint.h>

// Problem geometry (fixed by reference).
#define MM 512
#define NN 1024
#define IMG_SZ (MM*NN)
#define CHUNKW 64
#define OVL 10
#define TRIMW 5
#define ROWS 16     // row-tile per workgroup in chunk kernels
#define TW 80       // padded LDS row stride
#define COFF 8      // column offset inside LDS row (left halo pad)

#define LOG2E 1.4426950408889634f

typedef __attribute__((ext_vector_type(2))) float v2f;
typedef __attribute__((ext_vector_type(8))) float v8f;

__device__ __forceinline__ int iabs_(int a) { return a < 0 ? -a : a; }
__device__ __forceinline__ int iclamp_(int a, int lo, int hi) {
  return a < lo ? lo : (a > hi ? hi : a);
}

// fast base-2 exponential -> v_exp_f32
__device__ __forceinline__ float fexp2(float x) {
  return __builtin_amdgcn_exp2f(x);
}

// ---- CDNA5 async global->LDS copy (ASYNCcnt-tracked) --------------------
__device__ __forceinline__ void async_ld_b64(unsigned lds_off, const void* g) {
  // VDST = LDS byte address (VGPR), VADDR = 64-bit global address (VGPR pair)
  asm volatile("global_load_async_to_lds_b64 %0, %1, off"
               :: "v"(lds_off), "v"(g) : "memory");
}
__device__ __forceinline__ void wait_async0() {
  asm volatile("s_wait_asynccnt 0" ::: "memory");
}
__device__ __forceinline__ unsigned lds_off_of(const void* p) {
  // low 32 bits of a generic pointer into LDS == LDS byte offset
  return (unsigned)(uintptr_t)p;
}

// ---- min/max encoding for float atomics ---------------------------------
__device__ __forceinline__ unsigned encf(float f) {
  unsigned u = __float_as_uint(f);
  return (u & 0x80000000u) ? ~u : (u | 0x80000000u);
}
__device__ __forceinline__ float decf(unsigned e) {
  return (e & 0x80000000u) ? __uint_as_float(e ^ 0x80000000u)
                           : __uint_as_float(~e);
}

__global__ void init_stats_k(unsigned* stats) {
  stats[0] = 0xFFFFFFFFu;  // min (encoded)
  stats[1] = 0u;           // max (encoded)
}

__global__ __launch_bounds__(256) void reduce_minmax_k(
    const float* __restrict__ y, int n, unsigned* stats) {
  __shared__ unsigned smin[256], smax[256];
  unsigned mn = 0xFFFFFFFFu, mx = 0u;
  for (int i = blockIdx.x * 256 + threadIdx.x; i < n; i += gridDim.x * 256) {
    unsigned e = encf(y[i]);
    mn = mn < e ? mn : e;
    mx = mx > e ? mx : e;
  }
  smin[threadIdx.x] = mn; smax[threadIdx.x] = mx;
  __syncthreads();
  for (int st = 128; st > 0; st >>= 1) {
    if ((int)threadIdx.x < st) {
      smin[threadIdx.x] = min(smin[threadIdx.x], smin[threadIdx.x + st]);
      smax[threadIdx.x] = max(smax[threadIdx.x], smax[threadIdx.x + st]);
    }
    __syncthreads();
  }
  if (threadIdx.x == 0) {
    atomicMin(&stats[0], smin[0]);
    atomicMax(&stats[1], smax[0]);
  }
}

// ---- 5x5 box average as banded matmul via V_WMMA_F32_16X16X4_F32 --------
// Vertical pass: out[i,j] = sum_k Wv[i,k] * src[k,j], Wv banded (|k-i|<=2) = 0.2
// One wave per 16x16 tile; 6 K-tiles of 4 cover rows [r0-4, r0+20).
__global__ __launch_bounds__(32) void box5_v_k(
    const float* __restrict__ X, const float* __restrict__ Y,
    float* __restrict__ Xv, float* __restrict__ Yv) {
  const int lane = threadIdx.x;
  const int m = lane & 15, hi = lane >> 4;
  const int c0 = blockIdx.x * 16, r0 = blockIdx.y * 16;
  const float* src = blockIdx.z ? Y : X;
  float* dst = blockIdx.z ? Yv : Xv;
  v8f acc = {};
#pragma unroll
  for (int kt = 0; kt < 6; ++kt) {
    const int k0 = r0 - 4 + kt * 4;
    const int ka = k0 + 2 * hi, kb = ka + 1;  // A: K = {0,1}/{2,3} per half-wave
    v2f a, b;
    a.x = (ka >= 0 && ka < MM && iabs_(ka - (r0 + m)) <= 2) ? 0.2f : 0.0f;
    a.y = (kb >= 0 && kb < MM && iabs_(kb - (r0 + m)) <= 2) ? 0.2f : 0.0f;
    const int ra = iclamp_(ka, 0, MM - 1), rb = iclamp_(kb, 0, MM - 1);
    b.x = src[ra * NN + c0 + m];  // B rows K={0,2} in v0, {1,3} in v1
    b.y = src[rb * NN + c0 + m];
    acc = __builtin_amdgcn_wmma_f32_16x16x4_f32(false, a, false, b,
                                                (short)0, acc, false, false);
  }
#pragma unroll
  for (int v = 0; v < 8; ++v)
    dst[(r0 + v + hi * 8) * NN + c0 + m] = acc[v];
}

// Horizontal pass: out[i,j] = sum_k srcV[i,k] * Wh[k,j]; also det = orig - base.
__global__ __launch_bounds__(32) void box5_h_k(
    const float* __restrict__ X, const float* __restrict__ Y,
    const float* __restrict__ Xv, const float* __restrict__ Yv,
    float* __restrict__ Xb, float* __restrict__ Yb,
    float* __restrict__ Xd, float* __restrict__ Yd) {
  const int lane = threadIdx.x;
  const int m = lane & 15, hi = lane >> 4;
  const int c0 = blockIdx.x * 16, r0 = blockIdx.y * 16;
  const float* orig = blockIdx.z ? Y : X;
  const float* srcV = blockIdx.z ? Yv : Xv;
  float* dbase = blockIdx.z ? Yb : Xb;
  float* ddet  = blockIdx.z ? Yd : Xd;
  v8f acc = {};
#pragma unroll
  for (int kt = 0; kt < 6; ++kt) {
    const int k0 = c0 - 4 + kt * 4;
    const int ka = k0 + 2 * hi, kb = ka + 1;
    v2f a, b;
    const int ca = iclamp_(ka, 0, NN - 1), cb = iclamp_(kb, 0, NN - 1);
    a.x = srcV[(r0 + m) * NN + ca];
    a.y = srcV[(r0 + m) * NN + cb];
    b.x = (ka >= 0 && ka < NN && iabs_(ka - (c0 + m)) <= 2) ? 0.2f : 0.0f;
    b.y = (kb >= 0 && kb < NN && iabs_(kb - (c0 + m)) <= 2) ? 0.2f : 0.0f;
    acc = __builtin_amdgcn_wmma_f32_16x16x4_f32(false, a, false, b,
                                                (short)0, acc, false, false);
  }
#pragma unroll
  for (int v = 0; v < 8; ++v) {
    const int row = r0 + v + hi * 8, col = c0 + m;
    const float base = acc[v];
    dbase[row * NN + col] = base;
    ddet[row * NN + col]  = orig[row * NN + col] - base;
  }
}

// ---- shared tile staging: zero-fill, then async global->LDS --------------
// Zero-fill first (completes at the barrier), then async loads land on top.
template <int TR, int HALO>
__device__ __forceinline__ void stage_tiles(
    float* sx, float* sy, const float* __restrict__ gx,
    const float* __restrict__ gy, int grow0, int s, int cw) {
  // zero both tiles (8B stores; tile size is even and 16B-aligned)
  float2 z2; z2.x = 0.f; z2.y = 0.f;
  float2* zx = (float2*)sx;
  float2* zy = (float2*)sy;
  for (int i = threadIdx.x; i < (TR * TW) / 2; i += 256) { zx[i] = z2; zy[i] = z2; }
  __syncthreads();   // all ds-stores visible before async writes can land
  const int ppr = cw >> 1;                 // b64 pairs per row (cw is even)
  const int pc = threadIdx.x & 31;         // ppr <= 32 for this geometry
  const int pr0 = threadIdx.x >> 5;        // 8 row-walkers
  const unsigned bx = lds_off_of(sx), by = lds_off_of(sy);
  if (pc < ppr) {
    for (int pr = pr0; pr < TR; pr += 8) {
      const int gr = iclamp_(grow0 - HALO + pr, 0, MM - 1);
      const size_t goff = (size_t)gr * NN + s + 2 * pc;
      const unsigned loff = (unsigned)((pr * TW + COFF + 2 * pc) * 4);
      async_ld_b64(bx + loff, gx + goff);
      async_ld_b64(by + loff, gy + goff);
    }
  }
  wait_async0();
  __syncthreads();
}

// ---- 11x11 bilateral guided-filter chunk kernel -------------------------
__global__ __launch_bounds__(256) void base_chunk_k(
    const float* __restrict__ Xb, const float* __restrict__ Yb,
    const float* __restrict__ Xd,
    const unsigned* __restrict__ stats, const float* __restrict__ rp,
    float* __restrict__ out, int nch) {
  __shared__ __align__(16) float sx[(ROWS + 10) * TW];
  __shared__ __align__(16) float sy[(ROWS + 10) * TW];
  const int c = blockIdx.x, rt = blockIdx.y;
  const int s = c * (CHUNKW - OVL);
  const int e = (c == nch - 1) ? NN : ((s + CHUNKW < NN) ? s + CHUNKW : NN);
  const int cw = e - s;
  const int grow0 = rt * ROWS;

  const float sigma = rp[0] * (decf(stats[1]) - decf(stats[0]));
  const float nInvR = -(4.0f / (sigma * sigma)) * LOG2E;  // exp2-prescaled
  constexpr float invS = 1.0f / 915.0625f;                // 1/((11*11/4)^2)

  stage_tiles<ROWS + 10, 5>(sx, sy, Xb, Yb, grow0, s, cw);

  const int lo = (c == 0) ? 0 : TRIMW;
  const int hiC = (c == nch - 1) ? cw : cw - TRIMW;
  const int tcol = threadIdx.x & 63;
  const int trow = threadIdx.x >> 6;  // 4 row-walkers
  for (int col = tcol; col < cw; col += 64) {
    const bool owned = (col >= lo) && (col < hiC);
    for (int r = trow; r < ROWS; r += 4) {
      const int grow = grow0 + r;
      const float cen = sx[(r + 5) * TW + COFF + col];
      float wsum = 0.f, smy = 0.f, smx = 0.f, sxx = 0.f, sxy = 0.f;
#pragma unroll
      for (int dr = -5; dr <= 5; ++dr) {
        const bool rowOK = (unsigned)(grow + dr) < (unsigned)MM;
        const float* px = &sx[(r + 5 + dr) * TW + COFF + col];
        const float* py = &sy[(r + 5 + dr) * TW + COFF + col];
#pragma unroll
        for (int dc = -5; dc <= 5; ++dc) {
          const bool ok = rowOK && ((unsigned)(col + dc) < (unsigned)cw);
          const float v = px[dc];
          const float u = py[dc];
          const float d = v - cen;
          // w = exp2(-(d^2)/((sig/2)^2)*log2e - (dr^2+dc^2)/S^2*log2e)
          float w = fexp2(fmaf(d * d, nInvR,
                               -(float)(dr * dr + dc * dc) * invS * LOG2E));
          w = ok ? w : 0.0f;            // valid mask (reference ones-patch)
          const float wv = w * v;
          wsum += w;
          smy = fmaf(w, u, smy);
          smx += wv;
          sxx = fmaf(wv, v, sxx);
          sxy = fmaf(wv, u, sxy);
        }
      }
      const float inv = 1.0f / wsum;
      const float my = smy * inv, mx = smx * inv;
      const float vx = sxx * inv - mx * mx;
      const float cxy = sxy * inv - mx * my;
      const float A = cxy / (vx + 1e-6f);
      const float B = my - A * mx;
      if (owned) {
        const size_t g = (size_t)grow * NN + s + col;
        out[g] = A * cen + B + Xd[g];   // cen == Xbase at this pixel
      }
    }
  }
}

// ---- 5x5 bilateral detail chunk kernel: out += bd -----------------------
__global__ __launch_bounds__(256) void detail_chunk_k(
    const float* __restrict__ Xd, const float* __restrict__ Yd,
    const unsigned* __restrict__ stats, const float* __restrict__ rp,
    float* __restrict__ out, int nch) {
  __shared__ __align__(16) float sx[(ROWS + 4) * TW];
  __shared__ __align__(16) float sy[(ROWS + 4) * TW];
  const int c = blockIdx.x, rt = blockIdx.y;
  const int s = c * (CHUNKW - OVL);
  const int e = (c == nch - 1) ? NN : ((s + CHUNKW < NN) ? s + CHUNKW : NN);
  const int cw = e - s;
  const int grow0 = rt * ROWS;

  const float sigma = rp[0] * (decf(stats[1]) - decf(stats[0]));
  const float nInvR = -(4.0f / (sigma * sigma)) * LOG2E;
  constexpr float invS = 1.0f / 39.0625f;                 // 1/((5*5/4)^2)

  stage_tiles<ROWS + 4, 2>(sx, sy, Xd, Yd, grow0, s, cw);

  const int lo = (c == 0) ? 0 : TRIMW;
  const int hiC = (c == nch - 1) ? cw : cw - TRIMW;
  const int tcol = threadIdx.x & 63;
  const int trow = threadIdx.x >> 6;
  for (int col = tcol; col < cw; col += 64) {
    const bool owned = (col >= lo) && (col < hiC);
    for (int r = trow; r < ROWS; r += 4) {
      const int grow = grow0 + r;
      const float cen = sx[(r + 2) * TW + COFF + col];
      float wsum = 0.f, acc = 0.f;
#pragma unroll
      for (int dr = -2; dr <= 2; ++dr) {
        const bool rowOK = (unsigned)(grow + dr) < (unsigned)MM;
        const float* px = &sx[(r + 2 + dr) * TW + COFF + col];
        const float* py = &sy[(r + 2 + dr) * TW + COFF + col];
#pragma unroll
        for (int dc = -2; dc <= 2; ++dc) {
          const bool ok = rowOK && ((unsigned)(col + dc) < (unsigned)cw);
          const float v = px[dc];
          const float u = py[dc];
          const float d = v - cen;
          float w = fexp2(fmaf(d * d, nInvR,
                               -(float)(dr * dr + dc * dc) * invS * LOG2E));
          w = ok ? w : 0.0f;
          wsum += w;
          acc = fmaf(w, u - v, acc);
        }
      }
      if (owned) {
        const size_t g = (size_t)grow * NN + s + col;
        out[g] += acc / wsum;
      }
    }
  }
}

extern "C" void kernel_launch(void* const* d_in, const int* in_sizes, int n_in,
                              void* d_out, int out_size, void* d_ws, size_t ws_size,
                              hipStream_t stream) {
  const float* X = (const float*)d_in[0];
  const float* y = (const float*)d_in[1];
  const float* r = (const float*)d_in[2];
  float* out = (float*)d_out;

  unsigned* stats = (unsigned*)d_ws;
  float* base = (float*)d_ws + 64;
  float* Xv = base + 0 * (size_t)IMG_SZ;
  float* Yv = base + 1 * (size_t)IMG_SZ;
  float* Xb = base + 2 * (size_t)IMG_SZ;
  float* Yb = base + 3 * (size_t)IMG_SZ;
  float* Xd = base + 4 * (size_t)IMG_SZ;
  float* Yd = base + 5 * (size_t)IMG_SZ;

  // Replicate _chunks(N): starts step 54, last merged if width <= 5.
  int nch = 0;
  for (int s = 0; s < NN; s += (CHUNKW - OVL)) nch++;
  {
    int ls = (nch - 1) * (CHUNKW - OVL);
    int le = (ls + CHUNKW < NN) ? ls + CHUNKW : NN;
    if (le - ls <= 5) nch--;
  }

  init_stats_k<<<1, 1, 0, stream>>>(stats);
  reduce_minmax_k<<<512, 256, 0, stream>>>(y, IMG_SZ, stats);

  dim3 gb(NN / 16, MM / 16, 2);
  box5_v_k<<<gb, 32, 0, stream>>>(X, y, Xv, Yv);
  box5_h_k<<<gb, 32, 0, stream>>>(X, y, Xv, Yv, Xb, Yb, Xd, Yd);

  dim3 gc(nch, MM / ROWS);
  base_chunk_k<<<gc, 256, 0, stream>>>(Xb, Yb, Xd, stats, r, out, nch);
  detail_chunk_k<<<gc, 256, 0, stream>>>(Xd, Yd, stats, r, out, nch);
}